// HyperGrapyConv_77695958385168
// MI455X (gfx1250) — compile-verified
//
#include <hip/hip_runtime.h>
#include <stdint.h>

// ---------------------------------------------------------------------------
// HyperGraphConv: out = (x + 3*A*x)/4, A in COO (rows, cols, vals), d=128 f32.
//   Kernel 1 (hg_scale_init):  out = 0.25*x                    (dense, float4)
//   Kernel 2 (hg_spmm_scatter): out[row] += (0.75*val)*x[col]  (wave-per-nnz)
// CDNA5 specifics: wave32 => one 128-f32 row == 32 lanes * float4;
// async global->LDS staging of COO triples (ASYNCcnt, double buffered);
// global_prefetch of upcoming gather rows; native f32 L2 atomics.
// Working set (x 51MB + out 51MB + triples 38MB) is resident in the 192MB L2.
// ---------------------------------------------------------------------------

#define D_EMB 128
#define F4_PER_ROW (D_EMB / 4)      // 32 == lanes per wave
#define WAVE_SIZE 32
#define CHUNK 128                   // nnz triples per staged chunk (32 lanes * b128)
#define WAVES_PER_BLOCK 8
#define BLOCK_THREADS (WAVES_PER_BLOCK * WAVE_SIZE)
#define SCATTER_BLOCKS 1024
#define UNROLL 8

// Generic pointers into LDS carry the LDS byte offset in their low 32 bits
// (see ISA 10.2, LDS aperture addressing).
__device__ __forceinline__ uint32_t lds_offset(const void* p) {
  return (uint32_t)(uintptr_t)p;
}

// GLOBAL_LOAD_ASYNC_TO_LDS_B128: vdst = LDS byte address VGPR, vaddr = 64-bit
// global address VGPR pair, no SADDR ("off"). Tracked by ASYNCcnt.
__device__ __forceinline__ void async_load_b128_to_lds(void* lds_dst, const void* gsrc) {
  asm volatile("global_load_async_to_lds_b128 %0, %1, off"
               :
               : "v"(lds_offset(lds_dst)), "v"(gsrc)
               : "memory");
}

__device__ __forceinline__ void wait_async_le3() {
  asm volatile("s_wait_asynccnt 0x3" ::: "memory");
}
__device__ __forceinline__ void wait_async_le0() {
  asm volatile("s_wait_asynccnt 0x0" ::: "memory");
}

// One nnz handled by one full wave: lane l covers dims [4l, 4l+3].
// unsafeAtomicAdd -> global_atomic_add_f32 (no-return form, STOREcnt).
__device__ __forceinline__ void scatter_one(int row, int col, float v,
                                            const float4* __restrict__ x4,
                                            float* __restrict__ out, int lane) {
  float4 xv = x4[(size_t)col * F4_PER_ROW + lane];
  float* o = out + (size_t)row * D_EMB + lane * 4;
  unsafeAtomicAdd(o + 0, v * xv.x);
  unsafeAtomicAdd(o + 1, v * xv.y);
  unsafeAtomicAdd(o + 2, v * xv.z);
  unsafeAtomicAdd(o + 3, v * xv.w);
}

__global__ __launch_bounds__(BLOCK_THREADS) void hg_spmm_scatter(
    const int* __restrict__ rows, const int* __restrict__ cols,
    const float* __restrict__ vals, const float4* __restrict__ x4,
    float* __restrict__ out, long nnz) {
  // Per-wave private LDS slices (no cross-wave sharing -> no barriers):
  // [wave][2 buffers][rows CHUNK | cols CHUNK | vals CHUNK] = 24KB/block.
  __shared__ int smem[WAVES_PER_BLOCK * 2 * 3 * CHUNK];

  const int lane = threadIdx.x & (WAVE_SIZE - 1);
  const int wib = threadIdx.x >> 5;
  const long wave_id = (long)blockIdx.x * WAVES_PER_BLOCK + wib;
  const long nwaves = (long)gridDim.x * WAVES_PER_BLOCK;

  int* const wbase = smem + wib * (2 * 3 * CHUNK);
  const long nfull = nnz / CHUNK;

  auto stage = [&](int buf, long chunk) {
    int* dst = wbase + buf * (3 * CHUNK);
    const long base = chunk * CHUNK + (long)lane * 4;   // 16B-aligned
    async_load_b128_to_lds(dst + lane * 4,             rows + base);
    async_load_b128_to_lds(dst + CHUNK + lane * 4,     cols + base);
    async_load_b128_to_lds(dst + 2 * CHUNK + lane * 4, vals + base);
  };

  long c = wave_id;
  if (c < nfull) {
    int bi = 0;
    stage(0, c);
    for (;;) {
      const long nxt = c + nwaves;
      const bool more = (nxt < nfull);
      if (more) {
        stage(bi ^ 1, nxt);   // 3 new async ops behind the current 3
        wait_async_le3();     // async loads complete in order -> buf bi ready
      } else {
        wait_async_le0();
      }
      const int* s_rows = wbase + bi * (3 * CHUNK);
      const int* s_cols = s_rows + CHUNK;
      const float* s_vals = (const float*)(s_rows + 2 * CHUNK);
#pragma unroll UNROLL
      for (int e = 0; e < CHUNK; ++e) {
        if (e + UNROLL < CHUNK) {
          // All cols of the chunk already sit in LDS: prefetch gather row.
          int pcol = s_cols[e + UNROLL];
          __builtin_prefetch(&x4[(size_t)pcol * F4_PER_ROW + lane], 0, 3);
        }
        scatter_one(s_rows[e], s_cols[e], 0.75f * s_vals[e], x4, out, lane);
      }
      if (!more) break;
      c = nxt;
      bi ^= 1;
    }
  }

  // Tail (nnz % CHUNK != 0): wave 0 handles it straight from global.
  const long tbase = nfull * CHUNK;
  if (wave_id == 0) {
    for (long e = tbase; e < nnz; ++e) {
      scatter_one(rows[e], cols[e], 0.75f * vals[e], x4, out, lane);
    }
  }
}

// ---------------------------------------------------------------------------
__global__ __launch_bounds__(BLOCK_THREADS) void hg_scale_init(
    const float4* __restrict__ x4, float4* __restrict__ out4, long n4) {
  long i = (long)blockIdx.x * blockDim.x + threadIdx.x;
  const long stride = (long)gridDim.x * blockDim.x;
  for (; i < n4; i += stride) {
    float4 v = x4[i];
    float4 o;
    o.x = 0.25f * v.x;
    o.y = 0.25f * v.y;
    o.z = 0.25f * v.z;
    o.w = 0.25f * v.w;
    out4[i] = o;
  }
}

// ---------------------------------------------------------------------------
extern "C" void kernel_launch(void* const* d_in, const int* in_sizes, int n_in,
                              void* d_out, int out_size, void* d_ws, size_t ws_size,
                              hipStream_t stream) {
  const int* rows = (const int*)d_in[0];
  const int* cols = (const int*)d_in[1];
  const float* vals = (const float*)d_in[2];
  const float* x = (const float*)d_in[3];
  float* out = (float*)d_out;

  const long nnz = in_sizes[2];
  const long nelem = in_sizes[3];  // N_NODES * D_EMB
  const long n4 = nelem / 4;

  int blocks1 = (int)((n4 + BLOCK_THREADS - 1) / BLOCK_THREADS);
  if (blocks1 > 8192) blocks1 = 8192;
  hg_scale_init<<<blocks1, BLOCK_THREADS, 0, stream>>>(
      (const float4*)x, (float4*)out, n4);

  hg_spmm_scatter<<<SCATTER_BLOCKS, BLOCK_THREADS, 0, stream>>>(
      rows, cols, vals, (const float4*)x, out, nnz);
}